// SparseConv3d_29626684407885
// MI455X (gfx1250) — compile-verified
//
#include <hip/hip_runtime.h>

typedef float v2f __attribute__((ext_vector_type(2)));
typedef float v8f __attribute__((ext_vector_type(8)));

#define DD 112
#define HH 112
#define WW 112
#define CI 8
#define CO 8
#define NTAPS 27

#define ZD 3          // d halo rows
#define ZH 10         // 8 output rows + 2 halo
#define ZW 18         // 16 output cols + 2 halo
#define CP 10         // padded ci stride (8 used + 2 pad) -> conflict-free, b64-aligned

// Block = 8 wave32; wave w handles output row h0+w: a 16(w) x 8(oc) tile via
// 54x V_WMMA_F32_16X16X4_F32 (K = 8 ci x 27 taps). Input patch (zero-padded)
// and B fragments are staged in LDS so the hot loop is pure ds_load_b64+wmma.
__global__ __launch_bounds__(256) void SparseConv3d_wmma_f32_kernel(
    const float* __restrict__ x,     // [2][8][112][112][112]
    const float* __restrict__ wgt,   // [8][8][3][3][3]
    const float* __restrict__ bias,  // [8]
    float* __restrict__ out)         // [2][8][112][112][112]
{
  const int sH = WW;
  const int sD = HH * WW;
  const int sC = DD * HH * WW;
  const int sN = CI * sC;

  __shared__ float lx[ZD * ZH * ZW * CP];      // 5400 floats = 21.6 KB
  __shared__ float lw[NTAPS * 2 * 16 * 4];     // 3456 floats = 13.8 KB

  const int tid = threadIdx.x;

  int b = blockIdx.x;
  const int wq = b % 7;  b /= 7;               // w tile (16 wide)
  const int hb = b % 14; b /= 14;              // h block (8 rows)
  const int d  = b % DD; b /= DD;
  const int n  = b;
  const int h0 = hb * 8;
  const int w0 = wq * 16;

  // ---- stage zero-padded input patch: lx[zd][zh][w][ci] (ci stride CP) ----
  const float* xn = x + n * sN;
  for (int idx = tid; idx < ZD * ZH * CI * ZW; idx += 256) {
    const int zw_off = idx % ZW;
    int r = idx / ZW;
    const int ci = r % CI;     r /= CI;
    const int zh_off = r % ZH;
    const int zd_off = r / ZH;
    const int zd = d  + zd_off - 1;
    const int zh = h0 + zh_off - 1;
    const int zw = w0 + zw_off - 1;
    const bool ok = ((unsigned)zd < (unsigned)DD) &
                    ((unsigned)zh < (unsigned)HH) &
                    ((unsigned)zw < (unsigned)WW);
    const int zdc = ok ? zd : 0;
    const int zhc = ok ? zh : 0;
    const int zwc = ok ? zw : 0;
    // unconditional load from clamped (valid) address, masked by multiply
    const float v = xn[ci * sC + zdc * sD + zhc * sH + zwc] * (ok ? 1.0f : 0.0f);
    lx[((zd_off * ZH + zh_off) * ZW + zw_off) * CP + ci] = v;
  }

  // ---- stage B fragments: lw[t27][cig][n16][k4], zero for n>=8 ----
  for (int idx = tid; idx < NTAPS * 2 * 16 * 4; idx += 256) {
    const int k   = idx & 3;
    const int nn  = (idx >> 2) & 15;
    const int cig = (idx >> 6) & 1;
    const int t27 = idx >> 7;
    const int ci  = cig * 4 + k;
    const float v = wgt[((nn & 7) * CI + ci) * NTAPS + t27] *
                    (nn < CO ? 1.0f : 0.0f);
    lw[idx] = v;
  }

  __syncthreads();

  // ---- 54-WMMA branchless hot loop (EXEC all-ones) ----
  const int wave = tid >> 5;
  const int lane = tid & 31;
  const int m    = lane & 15;   // A: w offset (M); B: out channel (N)
  const int hi   = lane >> 4;   // K half: lanes 16-31 hold K=2,3

  // per-lane bases in v2f (8-byte) units; all 54 offsets are immediates
  const v2f* lxa = (const v2f*)lx + ((wave * ZW + m) * (CP / 2) + hi);
  const v2f* lwb = (const v2f*)lw + (m * 2 + hi);

  v8f acc = {0.f, 0.f, 0.f, 0.f, 0.f, 0.f, 0.f, 0.f};

#pragma unroll
  for (int kd = 0; kd < 3; ++kd) {
#pragma unroll
    for (int kh = 0; kh < 3; ++kh) {
#pragma unroll
      for (int kw = 0; kw < 3; ++kw) {
        const int t27 = (kd * 3 + kh) * 3 + kw;
#pragma unroll
        for (int cig = 0; cig < 2; ++cig) {
          // A: lx[kd][wave+kh][m+kw][cig*4 + hi*2 .. +1]
          const v2f A = lxa[((kd * ZH + kh) * ZW + kw) * (CP / 2) + cig * 2];
          // B: lw[t27][cig][m][hi*2 .. +1]
          const v2f B = lwb[(t27 * 2 + cig) * 32];
          acc = __builtin_amdgcn_wmma_f32_16x16x4_f32(
              false, A, false, B, (short)0, acc, false, false);
        }
      }
    }
  }

  // ---- store: VGPR j -> M = j + hi*8 (w), N = m (oc); valid lanes oc<8 ----
  const int oc = m;
  if (oc < CO) {
    const float bv = bias[oc];
    const int h = h0 + wave;
    float* op = out + (n * CO + oc) * sC + d * sD + h * sH + w0 + hi * 8;
#pragma unroll
    for (int j = 0; j < 8; ++j)
      __builtin_nontemporal_store(acc[j] + bv, op + j);
  }
}

extern "C" void kernel_launch(void* const* d_in, const int* in_sizes, int n_in,
                              void* d_out, int out_size, void* d_ws, size_t ws_size,
                              hipStream_t stream) {
  const float* x    = (const float*)d_in[0];
  const float* wgt  = (const float*)d_in[1];
  const float* bias = (const float*)d_in[2];
  float* out = (float*)d_out;

  const int blocks = 2 * DD * 14 * 7;          // 21,952 (exact)
  SparseConv3d_wmma_f32_kernel<<<blocks, 256, 0, stream>>>(x, wgt, bias, out);
}